// NodeAttention_67147518705696
// MI455X (gfx1250) — compile-verified
//
#include <hip/hip_runtime.h>
#include <hip/hip_bf16.h>

typedef float v2f __attribute__((ext_vector_type(2)));
typedef float v8f __attribute__((ext_vector_type(8)));

#define IN_CH 128
#define OUT_CH 128

// fp32 atomic add guaranteed to lower to the hardware global_atomic_add_f32
// (no-return form -> STOREcnt, no pre-op value round trip).
__device__ __forceinline__ void atomAddF32(float* p, float v) {
  unsafeAtomicAdd(p, v);
}

// ---------------------------------------------------------------- zero fill
__global__ void zero_f32(float* __restrict__ p, int n) {
  int i = blockIdx.x * blockDim.x + threadIdx.x;
  int stride = gridDim.x * blockDim.x;
  for (; i < n; i += stride) p[i] = 0.0f;
}

// ------------------------------------------------- w[n] = sigmoid(x[n].aw + b)
// one wave32 per node row; 4 loads/lane + shuffle reduction
__global__ void attn_kernel(const float* __restrict__ x,
                            const float* __restrict__ aw,
                            const float* __restrict__ ab,
                            float* __restrict__ w, int N) {
  int row  = (blockIdx.x * blockDim.x + threadIdx.x) >> 5;
  int lane = threadIdx.x & 31;
  if (row >= N) return;  // wave-uniform
  const float* xr = x + (size_t)row * IN_CH;
  float s = xr[lane      ] * aw[lane      ]
          + xr[lane + 32 ] * aw[lane + 32 ]
          + xr[lane + 64 ] * aw[lane + 64 ]
          + xr[lane + 96 ] * aw[lane + 96 ];
  #pragma unroll
  for (int off = 16; off > 0; off >>= 1) s += __shfl_down(s, off, 32);
  if (lane == 0) {
    float t = s + ab[0];
    w[row] = 1.0f / (1.0f + __expf(-t));
  }
}

// -------------------------------------- D[node] += w[edge]; Bcnt[edge] += 1
__global__ void degree_kernel(const int* __restrict__ nidx,
                              const int* __restrict__ eidx,
                              const float* __restrict__ w,
                              float* __restrict__ D, float* __restrict__ B,
                              int nnz) {
  int i = blockIdx.x * blockDim.x + threadIdx.x;
  if (i >= nnz) return;
  int n = nidx[i];
  int e = eidx[i];
  atomAddF32(&D[n], w[e]);
  atomAddF32(&B[e], 1.0f);
}

// --------------------------------------- v = (v > 0) ? 1/v : 0  (both arrays)
__global__ void recip_kernel(float* __restrict__ D, float* __restrict__ B, int n) {
  int i = blockIdx.x * blockDim.x + threadIdx.x;
  if (i >= n) return;
  float d = D[i]; D[i] = d > 0.0f ? 1.0f / d : 0.0f;
  float b = B[i]; B[i] = b > 0.0f ? 1.0f / b : 0.0f;
}

// ------------------------------------------------- xl = x @ lin_w^T via WMMA
// One wave computes a 16-row tile across all 128 output cols (8 col-tiles).
// A (16x4 f32) fragment per lane: float2 of x at row (tile*16 + lane%16),
// cols k + 2*(lane/16). B (4x16) fragment mirrors it on lin_w rows (out cols).
__global__ void __launch_bounds__(128)
gemm_xlw_kernel(const float* __restrict__ x,
                const float* __restrict__ lw,
                float* __restrict__ xl, int N) {
  int lane   = threadIdx.x & 31;
  int wave   = threadIdx.x >> 5;
  int tile   = blockIdx.x * 4 + wave;
  int ntiles = N >> 4;
  if (tile >= ntiles) return;  // wave-uniform: EXEC stays all-ones for WMMA

  int m     = lane & 15;
  int khalf = (lane >> 4) << 1;  // 0 for lanes 0-15, 2 for lanes 16-31

  const float* arow = x + ((size_t)(tile * 16 + m)) * IN_CH + khalf;

  v8f acc[8];
  #pragma unroll
  for (int ct = 0; ct < 8; ++ct) acc[ct] = (v8f){};

  for (int k = 0; k < IN_CH; k += 4) {
    v2f a = *(const v2f*)(arow + k);
    #pragma unroll
    for (int ct = 0; ct < 8; ++ct) {
      const float* brow = lw + ((size_t)(ct * 16 + m)) * IN_CH + khalf + k;
      v2f b = *(const v2f*)brow;
      acc[ct] = __builtin_amdgcn_wmma_f32_16x16x4_f32(
          /*neg_a=*/false, a, /*neg_b=*/false, b,
          /*c_mod=*/(short)0, acc[ct], /*reuse_a=*/false, /*reuse_b=*/false);
    }
  }

  // C/D layout: VGPR r -> row r (lanes 0-15) / row r+8 (lanes 16-31), col lane%16
  int rbase = tile * 16 + ((lane >> 4) << 3);
  #pragma unroll
  for (int ct = 0; ct < 8; ++ct) {
    int col = ct * 16 + m;
    #pragma unroll
    for (int r = 0; r < 8; ++r) {
      xl[(size_t)(rbase + r) * OUT_CH + col] = acc[ct][r];
    }
  }
}

// --------------------- dst[didx[i]] += src[sidx[i]]  (row scatter, wave/nnz)
__global__ void scatter_rows_kernel(const float* __restrict__ src,
                                    const int* __restrict__ sidx,
                                    float* __restrict__ dst,
                                    const int* __restrict__ didx,
                                    int nnz) {
  int wid  = (blockIdx.x * blockDim.x + threadIdx.x) >> 5;
  int lane = threadIdx.x & 31;
  if (wid >= nnz) return;  // wave-uniform
  // wid is wave-uniform: force scalar index loads via readfirstlane
  int s = __builtin_amdgcn_readfirstlane(sidx[wid]);
  int d = __builtin_amdgcn_readfirstlane(didx[wid]);
  const float4 v = *(const float4*)(src + (size_t)s * OUT_CH + lane * 4);
  float* p = dst + (size_t)d * OUT_CH + lane * 4;
  atomAddF32(p + 0, v.x);
  atomAddF32(p + 1, v.y);
  atomAddF32(p + 2, v.z);
  atomAddF32(p + 3, v.w);
}

// ------------------------------------------------ p[i] *= s[row(i)]
__global__ void scale_rows_kernel(float* __restrict__ p,
                                  const float* __restrict__ s, int n) {
  int i = blockIdx.x * blockDim.x + threadIdx.x;
  if (i < n) p[i] *= s[i >> 7];
}

// ------------------------------------- out[i] = out[i]*Dinv[row] + bias[col]
__global__ void finalize_kernel(float* __restrict__ out,
                                const float* __restrict__ Dinv,
                                const float* __restrict__ bias, int n) {
  int i = blockIdx.x * blockDim.x + threadIdx.x;
  if (i < n) out[i] = out[i] * Dinv[i >> 7] + bias[i & 127];
}

// ---------------------------------------------------------------------------
extern "C" void kernel_launch(void* const* d_in, const int* in_sizes, int n_in,
                              void* d_out, int out_size, void* d_ws, size_t ws_size,
                              hipStream_t stream) {
  const float* x      = (const float*)d_in[0];
  const int*   hidx   = (const int*)d_in[1];
  const float* attn_w = (const float*)d_in[2];
  const float* attn_b = (const float*)d_in[3];
  const float* lin_w  = (const float*)d_in[4];
  const float* bias   = (const float*)d_in[5];
  float*       out    = (float*)d_out;

  const int N   = in_sizes[0] / IN_CH;  // 50000 nodes
  const int M   = N;                    // N_EDGES == N_NODES
  const int nnz = in_sizes[1] / 2;      // 800000
  const int* nidx = hidx;          // row 0: node ids
  const int* eidx = hidx + nnz;    // row 1: hyperedge ids

  // workspace layout (floats): w[N] | Dinv[N] | Binv[M] | xl[N*128] | efeat[M*128]
  float* ws    = (float*)d_ws;
  float* w     = ws;
  float* Dinv  = w + N;
  float* Binv  = Dinv + N;
  float* xl    = Binv + M;
  float* efeat = xl + (size_t)N * OUT_CH;

  // zero the accumulated buffers (Dinv/Binv contiguous)
  zero_f32<<<256, 256, 0, stream>>>(Dinv, N + M);
  zero_f32<<<2048, 256, 0, stream>>>(efeat, M * OUT_CH);
  zero_f32<<<2048, 256, 0, stream>>>(out, N * OUT_CH);

  // per-node attention score -> per-edge weight
  attn_kernel<<<(N + 7) / 8, 256, 0, stream>>>(x, attn_w, attn_b, w, N);

  // degrees
  degree_kernel<<<(nnz + 255) / 256, 256, 0, stream>>>(nidx, eidx, w, Dinv, Binv, nnz);
  recip_kernel<<<(N + 255) / 256, 256, 0, stream>>>(Dinv, Binv, N);

  // xl = x @ lin_w^T   (fp32 WMMA, 4 waves of row-tiles per block)
  int ntiles = N >> 4;
  gemm_xlw_kernel<<<(ntiles + 3) / 4, 128, 0, stream>>>(x, lin_w, xl, N);

  // node -> edge aggregation (unscaled), then scale by Binv
  scatter_rows_kernel<<<(nnz * 8 + 63) / 64, 256, 0, stream>>>(xl, nidx, efeat, eidx, nnz);
  scale_rows_kernel<<<(M * OUT_CH + 255) / 256, 256, 0, stream>>>(efeat, Binv, M * OUT_CH);

  // edge -> node aggregation, then out = out*Dinv + bias
  scatter_rows_kernel<<<(nnz * 8 + 63) / 64, 256, 0, stream>>>(efeat, eidx, out, nidx, nnz);
  finalize_kernel<<<(N * OUT_CH + 255) / 256, 256, 0, stream>>>(out, Dinv, bias, N * OUT_CH);
}